// WeightedGraphConv_40441412059453
// MI455X (gfx1250) — compile-verified
//
#include <hip/hip_runtime.h>

#define T_DIM 2
#define F_DIM 128   // F_IN == F_OUT == 128

typedef float v2f __attribute__((ext_vector_type(2)));
typedef float v8f __attribute__((ext_vector_type(8)));

// ---------------------------------------------------------------------------
// Kernel 1: zero the h accumulator (d_ws is poisoned 0xAA, not re-zeroed).
// ---------------------------------------------------------------------------
__global__ void __launch_bounds__(256) wgc_zero(float4* __restrict__ p, int n4) {
  int i = blockIdx.x * 256 + threadIdx.x;
  if (i < n4) p[i] = make_float4(0.f, 0.f, 0.f, 0.f);
}

// ---------------------------------------------------------------------------
// Kernel 2: edge scatter.  One wave32 per edge; lane l handles float4 chunk l
// of the 128-float feature row for each t.  node_features (51.2MB) and h
// (51.2MB) both fit in the 192MB L2, so gathers and atomics stay on-chip.
// ---------------------------------------------------------------------------
__global__ void __launch_bounds__(256) wgc_scatter(
    const float* __restrict__ nf,   // [N, T, F]
    const float* __restrict__ ew,   // [T, E]
    const int*   __restrict__ src,  // [E]
    const int*   __restrict__ dst,  // [E]
    float*       __restrict__ h,    // [N, T, F] accumulator
    int E) {
  int e = blockIdx.x * 8 + (threadIdx.x >> 5);   // wave-uniform edge id
  if (e >= E) return;
  int lane = threadIdx.x & 31;

  int   s  = src[e];
  int   d  = dst[e];
  float w0 = ew[e];        // t = 0
  float w1 = ew[E + e];    // t = 1

  const float4* x = (const float4*)(nf + (size_t)s * (T_DIM * F_DIM));
  float4 v0 = x[lane];        // t=0, floats [4*lane, 4*lane+4)
  float4 v1 = x[32 + lane];   // t=1

  float* hd = h + (size_t)d * (T_DIM * F_DIM);
  int o = 4 * lane;
  atomicAdd(hd + o + 0, v0.x * w0);
  atomicAdd(hd + o + 1, v0.y * w0);
  atomicAdd(hd + o + 2, v0.z * w0);
  atomicAdd(hd + o + 3, v0.w * w0);
  atomicAdd(hd + F_DIM + o + 0, v1.x * w1);
  atomicAdd(hd + F_DIM + o + 1, v1.y * w1);
  atomicAdd(hd + F_DIM + o + 2, v1.z * w1);
  atomicAdd(hd + F_DIM + o + 3, v1.w * w1);
}

// ---------------------------------------------------------------------------
// Kernel 3: out[m, o] = sum_f h[m, f] * W[o, f] + b[o]   (m over N*T rows)
// Exact-fp32 GEMM via V_WMMA_F32_16X16X4_F32.  Block = 8 waves; block owns a
// 16-row stripe, wave w owns 16-column tile w.  Two accumulator chains for
// pipeline overlap; bias folded into c0 init.  No divergence -> EXEC all 1s.
// ---------------------------------------------------------------------------
__global__ void __launch_bounds__(256) wgc_gemm(
    const float* __restrict__ h,     // [rows, 128]
    const float* __restrict__ W,     // [128, 128] (row = output feature)
    const float* __restrict__ bias,  // [128]
    float*       __restrict__ out,   // [rows, 128]
    int rows) {
  const int wave = threadIdx.x >> 5;
  const int lane = threadIdx.x & 31;
  const int half = lane >> 4;        // 0: lanes 0-15, 1: lanes 16-31
  const int idx  = lane & 15;
  const int row_base = blockIdx.x * 16;
  const int col_base = wave * 16;

  // A[idx][k] with k = kk + 2*half + {0,1}  (16x4 f32 A layout)
  const float* ap = h + (size_t)(row_base + idx) * F_DIM + 2 * half;
  // B[k][idx] = W[col_base+idx][k]          (4x16 f32 B layout, mirrored)
  const float* bp = W + (size_t)(col_base + idx) * F_DIM + 2 * half;

  float bv = bias[col_base + idx];
  v8f c0, c1;
#pragma unroll
  for (int r = 0; r < 8; ++r) { c0[r] = bv; c1[r] = 0.f; }

#pragma unroll
  for (int kk = 0; kk < F_DIM; kk += 8) {
    v2f a0 = *(const v2f*)(ap + kk);
    v2f b0 = *(const v2f*)(bp + kk);
    v2f a1 = *(const v2f*)(ap + kk + 4);
    v2f b1 = *(const v2f*)(bp + kk + 4);
    c0 = __builtin_amdgcn_wmma_f32_16x16x4_f32(
        false, a0, false, b0, (short)0, c0, false, false);
    c1 = __builtin_amdgcn_wmma_f32_16x16x4_f32(
        false, a1, false, b1, (short)0, c1, false, false);
  }

  // C/D layout: VGPR r, lane 16*half+idx  ->  element (M = r + 8*half, N = idx)
  float* op = out + (size_t)(row_base + 8 * half) * F_DIM + col_base + idx;
#pragma unroll
  for (int r = 0; r < 8; ++r) op[(size_t)r * F_DIM] = c0[r] + c1[r];
}

// ---------------------------------------------------------------------------
extern "C" void kernel_launch(void* const* d_in, const int* in_sizes, int n_in,
                              void* d_out, int out_size, void* d_ws, size_t ws_size,
                              hipStream_t stream) {
  const float* nf  = (const float*)d_in[0];  // node_features [N,T,F]
  const float* ew  = (const float*)d_in[1];  // edge_weights  [T,E]
  const int*   src = (const int*)  d_in[2];  // [E]
  const int*   dst = (const int*)  d_in[3];  // [E]
  const float* W   = (const float*)d_in[4];  // [F_OUT,F_IN]
  const float* b   = (const float*)d_in[5];  // [F_OUT]
  float*       out = (float*)d_out;

  const int N = in_sizes[0] / (T_DIM * F_DIM);
  const int E = in_sizes[2];
  float* h = (float*)d_ws;                   // [N,T,F] fp32 = 51.2 MB

  // 1) zero accumulator
  int n4 = (N * T_DIM * F_DIM) / 4;
  wgc_zero<<<(n4 + 255) / 256, 256, 0, stream>>>((float4*)h, n4);

  // 2) weighted scatter-add over edges (8 edges per 256-thread block)
  wgc_scatter<<<(E + 7) / 8, 256, 0, stream>>>(nf, ew, src, dst, h, E);

  // 3) fp32 WMMA GEMM + bias  (rows = N*T = 100000 = 6250 * 16)
  int rows = N * T_DIM;
  wgc_gemm<<<rows / 16, 256, 0, stream>>>(h, W, b, out, rows);
}